// Decoder_14422500179981
// MI455X (gfx1250) — compile-verified
//
#include <hip/hip_runtime.h>
#include <hip/hip_bf16.h>
#include <math.h>

// Problem constants (from the reference)
#define BB 32
#define SS 48
#define LL 48
#define HH 1024
#define EE 1024
#define VV 32000
#define H3 3072

typedef float v2f __attribute__((ext_vector_type(2)));
typedef float v8f __attribute__((ext_vector_type(8)));

// ---------------------------------------------------------------------------
// Per-wave fp32 WMMA GEMM: computes C[0:32, n0:n0+16] = act(A(32xK) @ W + bias)
// A is read as two row-major segments (A0 for k<split, A1 for k>=split), both
// with leading dim lda. Wc/biasc/Cc are already offset to column n0.
// Uses V_WMMA_F32_16X16X4_F32 (native fp32 matrix core path on gfx1250).
// ---------------------------------------------------------------------------
__device__ __forceinline__ void wave_gemm32(
    const float* __restrict__ A0, const float* __restrict__ A1,
    int split, int lda,
    const float* __restrict__ Wc, long ldw,
    const float* __restrict__ biasc,
    float* __restrict__ Cc, long ldc,
    int K, bool do_relu)
{
    const int lane = threadIdx.x & 31;
    const int r    = lane & 15;   // A row (0-15) / B,C column within tile
    const int kg   = lane >> 4;   // K-pair group select

    v8f acc0 = {}; // rows 0..15
    v8f acc1 = {}; // rows 16..31

    for (int k = 0; k < K; k += 4) {
        const int ka = k + 2 * kg;          // even, so float2 loads are aligned
        v2f a0, a1, b;
        if (ka < split) {
            a0 = *(const v2f*)(A0 + (long)r        * lda + ka);
            a1 = *(const v2f*)(A0 + (long)(r + 16) * lda + ka);
        } else {
            const int kb = ka - split;
            a0 = *(const v2f*)(A1 + (long)r        * lda + kb);
            a1 = *(const v2f*)(A1 + (long)(r + 16) * lda + kb);
        }
        b.x = Wc[(long)ka       * ldw + r];
        b.y = Wc[(long)(ka + 1) * ldw + r];
        acc0 = __builtin_amdgcn_wmma_f32_16x16x4_f32(
                   false, a0, false, b, (short)0, acc0, false, false);
        acc1 = __builtin_amdgcn_wmma_f32_16x16x4_f32(
                   false, a1, false, b, (short)0, acc1, false, false);
    }

    const float bv = biasc[r];
#pragma unroll
    for (int i = 0; i < 8; ++i) {
        const int row = i + 8 * kg;
        float v0 = acc0[i] + bv;
        float v1 = acc1[i] + bv;
        if (do_relu) { v0 = fmaxf(v0, 0.f); v1 = fmaxf(v1, 0.f); }
        Cc[(long)row        * ldc + r] = v0;
        Cc[(long)(row + 16) * ldc + r] = v1;
    }
}

// ---------------------------------------------------------------------------
// Kernel 0: xs[s][b][:] = relu(emb[label[b][s]])   (all steps at once)
// ---------------------------------------------------------------------------
__global__ void emb_relu_kernel(const int* __restrict__ label,
                                const float* __restrict__ emb,
                                float* __restrict__ xs)
{
    const int sb = blockIdx.x;            // sb = s*B + b
    const int s  = sb / BB;
    const int b  = sb % BB;
    const int tok = label[b * SS + s];
    const float* src = emb + (long)tok * EE;
    float* dst = xs + (long)sb * EE;
    for (int i = threadIdx.x; i < EE; i += blockDim.x)
        dst[i] = fmaxf(src[i], 0.f);
}

// ---------------------------------------------------------------------------
// Kernel 1: sim[:,l] = cat(xe,h) @ attn_W[:,l] + attn_b[l]; softmax over the
// BATCH axis (faithful to reference's dim=0 softmax); one block per l.
// ---------------------------------------------------------------------------
__global__ void attn_kernel(const float* __restrict__ xe,
                            const float* __restrict__ h,
                            const float* __restrict__ attn_W,
                            const float* __restrict__ attn_b,
                            float* __restrict__ wout)
{
    const int l   = blockIdx.x;     // 0..L-1
    const int tid = threadIdx.x;    // 0..255
    const int b   = tid >> 3;       // 8 threads per batch row
    const int sub = tid & 7;

    float acc = 0.f;
    for (int k = sub; k < 2 * HH; k += 8) {
        const float av = (k < HH) ? xe[b * HH + k] : h[b * HH + (k - HH)];
        acc += av * attn_W[(long)k * LL + l];
    }
    __shared__ float red[256];
    __shared__ float sim[BB];
    red[tid] = acc;
    __syncthreads();
    if (sub < 4) red[tid] += red[tid + 4];
    __syncthreads();
    if (sub < 2) red[tid] += red[tid + 2];
    __syncthreads();
    if (sub == 0) sim[b] = red[tid] + red[tid + 1] + attn_b[l];
    __syncthreads();
    if (tid == 0) {
        float m = sim[0];
        for (int i = 1; i < BB; ++i) m = fmaxf(m, sim[i]);
        float ssum = 0.f;
        for (int i = 0; i < BB; ++i) { sim[i] = __expf(sim[i] - m); ssum += sim[i]; }
        const float inv = 1.f / ssum;
        for (int i = 0; i < BB; ++i) wout[i * LL + l] = sim[i] * inv;
    }
}

// ---------------------------------------------------------------------------
// Kernel 2: ctx[b,h] = sum_l w[b,l] * enc[b,l,h]; one block per batch row.
// ---------------------------------------------------------------------------
__global__ void ctx_kernel(const float* __restrict__ w,
                           const float* __restrict__ enc,
                           float* __restrict__ ctx)
{
    const int b = blockIdx.x;
    for (int hh = threadIdx.x; hh < HH; hh += blockDim.x) {
        float acc = 0.f;
        for (int l = 0; l < LL; ++l)
            acc += w[b * LL + l] * enc[((long)(b * LL + l)) * HH + hh];
        ctx[b * HH + hh] = acc;
    }
}

// ---------------------------------------------------------------------------
// Kernel 3: g = relu(cat(ctx, xe) @ comb_W + comb_b)   (32x2048 @ 2048x1024)
// 8 waves/block, one 32x16 output tile per wave. N=1024 -> 8 blocks.
// ---------------------------------------------------------------------------
__global__ void comb_kernel(const float* __restrict__ ctx,
                            const float* __restrict__ xe,
                            const float* __restrict__ comb_W,
                            const float* __restrict__ comb_b,
                            float* __restrict__ g)
{
    const int tile = blockIdx.x * 8 + (threadIdx.x >> 5);
    const int n0   = tile * 16;
    wave_gemm32(ctx, xe, HH, HH,
                comb_W + n0, HH, comb_b + n0,
                g + n0, HH, 2 * HH, /*relu=*/true);
}

// ---------------------------------------------------------------------------
// Kernel 4: fused gate GEMMs. gates[:,0:3072] = g @ W_ih + b_ih,
//           gates[:,3072:6144] = h @ W_hh + b_hh.  Wave-uniform select.
//           N_total = 6144 -> 48 blocks of 8 waves.
// ---------------------------------------------------------------------------
__global__ void gates_kernel(const float* __restrict__ g,
                             const float* __restrict__ h,
                             const float* __restrict__ W_ih,
                             const float* __restrict__ W_hh,
                             const float* __restrict__ b_ih,
                             const float* __restrict__ b_hh,
                             float* __restrict__ gates)
{
    const int tile = blockIdx.x * 8 + (threadIdx.x >> 5);
    const int n0   = tile * 16;
    if (n0 < H3) {
        wave_gemm32(g, g, HH, HH,
                    W_ih + n0, H3, b_ih + n0,
                    gates + n0, 2 * H3, HH, /*relu=*/false);
    } else {
        const int wn0 = n0 - H3;
        wave_gemm32(h, h, HH, HH,
                    W_hh + wn0, H3, b_hh + wn0,
                    gates + n0, 2 * H3, HH, /*relu=*/false);
    }
}

// ---------------------------------------------------------------------------
// Kernel 5: GRU elementwise (torch gate order r, z, n).
// ---------------------------------------------------------------------------
__global__ void gru_kernel(const float* __restrict__ gates,
                           const float* __restrict__ hprev,
                           float* __restrict__ hnext)
{
    const int idx = blockIdx.x * blockDim.x + threadIdx.x;  // 0..B*H-1
    const int b = idx >> 10;
    const int j = idx & (HH - 1);
    const float* gx = gates + (long)b * (2 * H3);
    const float* gh = gx + H3;
    const float r = 1.f / (1.f + __expf(-(gx[j]          + gh[j])));
    const float z = 1.f / (1.f + __expf(-(gx[j + HH]     + gh[j + HH])));
    const float n = tanhf(gx[j + 2 * HH] + r * gh[j + 2 * HH]);
    hnext[idx] = (1.f - z) * n + z * hprev[idx];
}

// ---------------------------------------------------------------------------
// Kernel 6: logits[b, s, :] = h_new @ out_W + out_b  (32x1024 @ 1024x32000)
// N = 32000 -> 2000 tiles -> 250 blocks of 8 waves. Writes into d_out slice.
// ---------------------------------------------------------------------------
__global__ void out_gemm_kernel(const float* __restrict__ h,
                                const float* __restrict__ out_W,
                                const float* __restrict__ out_b,
                                float* __restrict__ out, int s)
{
    const int tile = blockIdx.x * 8 + (threadIdx.x >> 5);
    const int n0   = tile * 16;
    wave_gemm32(h, h, HH, HH,
                out_W + n0, VV, out_b + n0,
                out + (long)s * VV + n0, (long)SS * VV, HH, /*relu=*/false);
}

// ---------------------------------------------------------------------------
// Kernel 7: in-place log_softmax over V for row (b, s): online max/sum-exp
// per thread, LDS pairwise (m,s) reduction, then subtract lse.
// ---------------------------------------------------------------------------
__global__ void logsoftmax_kernel(float* __restrict__ out, int s)
{
    const int b   = blockIdx.x;
    const int tid = threadIdx.x;
    float* row = out + ((long)b * SS + s) * VV;

    float m = -INFINITY, sum = 0.f;
    for (int v = tid; v < VV; v += blockDim.x) {
        const float x = row[v];
        if (x > m) { sum = sum * __expf(m - x) + 1.f; m = x; }
        else       { sum += __expf(x - m); }
    }
    __shared__ float ms[256], ss[256];
    ms[tid] = m; ss[tid] = sum;
    __syncthreads();
    for (int off = 128; off > 0; off >>= 1) {
        if (tid < off) {
            const float m2 = ms[tid + off], s2 = ss[tid + off];
            const float M  = fmaxf(ms[tid], m2);
            ss[tid] = ss[tid] * __expf(ms[tid] - M) + s2 * __expf(m2 - M);
            ms[tid] = M;
        }
        __syncthreads();
    }
    const float lse = ms[0] + __logf(ss[0]);
    for (int v = tid; v < VV; v += blockDim.x)
        row[v] -= lse;
}

// ---------------------------------------------------------------------------
extern "C" void kernel_launch(void* const* d_in, const int* in_sizes, int n_in,
                              void* d_out, int out_size, void* d_ws, size_t ws_size,
                              hipStream_t stream)
{
    (void)in_sizes; (void)n_in; (void)out_size; (void)ws_size;

    const int*   label   = (const int*)  d_in[0];
    const float* init_h  = (const float*)d_in[1];   // (1, B, H)
    const float* enc     = (const float*)d_in[2];   // (B, L, H)
    const float* emb     = (const float*)d_in[3];   // (V, E)
    const float* attn_W  = (const float*)d_in[4];   // (2H, L)
    const float* attn_b  = (const float*)d_in[5];   // (L,)
    const float* comb_W  = (const float*)d_in[6];   // (2H, H)
    const float* comb_b  = (const float*)d_in[7];   // (H,)
    const float* W_ih    = (const float*)d_in[8];   // (E, 3H)
    const float* W_hh    = (const float*)d_in[9];   // (H, 3H)
    const float* b_ih    = (const float*)d_in[10];  // (3H,)
    const float* b_hh    = (const float*)d_in[11];  // (3H,)
    const float* out_W   = (const float*)d_in[12];  // (H, V)
    const float* out_b   = (const float*)d_in[13];  // (V,)
    float* out = (float*)d_out;                     // (B, S, V)

    // Workspace layout (floats)
    float* ws    = (float*)d_ws;
    float* xs    = ws;                                  // S*B*E
    float* h0    = xs   + (long)SS * BB * EE;           // B*H
    float* h1    = h0   + BB * HH;                      // B*H
    float* wat   = h1   + BB * HH;                      // B*L
    float* ctx   = wat  + BB * LL;                      // B*H
    float* g     = ctx  + BB * HH;                      // B*H
    float* gates = g    + BB * HH;                      // B*6144

    // h0 = init_hidden[0]
    hipMemcpyAsync(h0, init_h, (size_t)BB * HH * sizeof(float),
                   hipMemcpyDeviceToDevice, stream);

    // All-step embedding + relu
    emb_relu_kernel<<<SS * BB, 256, 0, stream>>>(label, emb, xs);

    for (int s = 0; s < SS; ++s) {
        const float* hprev = (s & 1) ? h1 : h0;
        float*       hnext = (s & 1) ? h0 : h1;
        const float* xe = xs + (long)s * BB * EE;

        attn_kernel<<<LL, 256, 0, stream>>>(xe, hprev, attn_W, attn_b, wat);
        ctx_kernel<<<BB, 256, 0, stream>>>(wat, enc, ctx);
        comb_kernel<<<HH / 128, 256, 0, stream>>>(ctx, xe, comb_W, comb_b, g);
        gates_kernel<<<(2 * H3) / 128, 256, 0, stream>>>(g, hprev, W_ih, W_hh,
                                                         b_ih, b_hh, gates);
        gru_kernel<<<(BB * HH) / 256, 256, 0, stream>>>(gates, hprev, hnext);
        out_gemm_kernel<<<VV / 128, 256, 0, stream>>>(hnext, out_W, out_b, out, s);
        logsoftmax_kernel<<<BB, 256, 0, stream>>>(out, s);
    }
}